// OneHotSlicer_51238959841479
// MI455X (gfx1250) — compile-verified
//
#include <hip/hip_runtime.h>
#include <cstdint>

// Problem constants (from the reference): x is (128, 4, 2048) f32, K=200.
#define B_      128
#define C_      4
#define L_      2048
#define K_      200
#define LP_     1849          // L - K + 1
#define TILE_P  128           // windows per block
#define TILES   15            // ceil(LP_ / TILE_P)
#define SEG     328           // TILE_P - 1 + K_ = 327, padded to multiple of 4
#define SEG4    82            // SEG / 4  (float4s per channel segment)
#define NTHREADS 256

typedef __attribute__((address_space(3))) float lds_f32;
typedef __attribute__((ext_vector_type(4))) float v4f;   // true vector type for NT store

__global__ __launch_bounds__(NTHREADS)
void slicer_im2col_kernel(const float* __restrict__ x, float* __restrict__ out) {
    __shared__ float smem[C_ * SEG];   // 4 * 328 * 4B = 5248 B

    const int tile = blockIdx.x;       // 0..14
    const int b    = blockIdx.y;       // 0..127
    const int p0   = tile * TILE_P;
    const int tid  = threadIdx.x;

    // ---------------- Phase 1: async global -> LDS segment fill ----------------
    // Copy x[b, c, p0 .. p0+SEG) for all channels into LDS with
    // global_load_async_to_lds_b128 (ASYNCcnt-tracked, no VGPR data path).
    const int avail4 = (L_ - p0) >> 2;               // float4s left in the row
    for (int t = tid; t < C_ * SEG4; t += NTHREADS) {
        int c = t / SEG4;
        int j = t % SEG4;
        if (j >= avail4) j = avail4 - 1;             // clamp at row end (dup-safe)
        unsigned goff = (unsigned)((((b * C_ + c) * L_) + p0 + 4 * j) * 4); // bytes
        unsigned lds_byte =
            (unsigned)(uintptr_t)(lds_f32*)(&smem[c * SEG + 4 * j]);        // LDS byte addr
        asm volatile("global_load_async_to_lds_b128 %0, %1, %2"
                     :
                     : "v"(lds_byte), "v"(goff), "s"(x)
                     : "memory");
    }
    asm volatile("s_wait_asynccnt 0" ::: "memory");  // own wave's async copies done
    __syncthreads();                                 // all waves' copies visible

    // ---------------- Phase 2: coalesced 128-bit window stores ----------------
    // out4 index = (b*Lp + p0)*200 + t with t = (pl*4 + c)*50 + f4, so consecutive
    // lanes store consecutive float4s -> fully contiguous per-block output region.
    const int valid = (LP_ - p0 < TILE_P) ? (LP_ - p0) : TILE_P;
    v4f* out4 = (v4f*)out;
    const long long base4 = (long long)(b * LP_ + p0) * (long long)(C_ * K_ / 4);

    for (int t = tid; t < TILE_P * C_ * (K_ / 4); t += NTHREADS) {
        int f4 = t % (K_ / 4);          // 0..49  -> k = 4*f4
        int c  = (t / (K_ / 4)) % C_;   // channel
        int pl = t / (C_ * (K_ / 4));   // local window
        if (pl < valid) {
            const float* s = &smem[c * SEG + pl + 4 * f4];
            v4f v = { s[0], s[1], s[2], s[3] };
            __builtin_nontemporal_store(v, &out4[base4 + (long long)t]);
        }
    }
}

extern "C" void kernel_launch(void* const* d_in, const int* in_sizes, int n_in,
                              void* d_out, int out_size, void* d_ws, size_t ws_size,
                              hipStream_t stream) {
    const float* x = (const float*)d_in[0];
    float* out = (float*)d_out;
    (void)in_sizes; (void)n_in; (void)out_size; (void)d_ws; (void)ws_size;

    dim3 grid(TILES, B_, 1);
    dim3 block(NTHREADS, 1, 1);
    slicer_im2col_kernel<<<grid, block, 0, stream>>>(x, out);
}